// Aggregate_34840774705429
// MI455X (gfx1250) — compile-verified
//
#include <hip/hip_runtime.h>
#include <stdint.h>

typedef __attribute__((ext_vector_type(4))) float v4f;

#define BDIM 16
#define NDIM 8192
#define KDIM 32
#define DDIM 64
#define NSEG 64
#define NPB  (NDIM / NSEG)   // 128 n-values per block
#define THREADS 512

__global__ __launch_bounds__(THREADS) void aggregate_kernel(
    const float* __restrict__ A, const float* __restrict__ R,
    float* __restrict__ out) {
  // A tile for this block: NPB x KDIM floats = 16 KB (contiguous in global)
  __shared__ float sA[NPB * KDIM];

  const int b   = blockIdx.x / NSEG;
  const int seg = blockIdx.x % NSEG;
  const int n0  = seg * NPB;
  const int t   = threadIdx.x;

  // ---- Stage A tile into LDS via CDNA5 async global->LDS copy (ASYNCcnt) ----
  {
    const float* aseg = A + ((size_t)b * NDIM + n0) * KDIM;  // 16 KB contiguous
    unsigned long long gbase = (unsigned long long)(uintptr_t)aseg;
    unsigned lds_base = (unsigned)(uintptr_t)(void*)sA;      // low 32 bits = LDS offset
    unsigned off0 = (unsigned)t * 16u;                        // 512 lanes x 16 B = 8 KB
    unsigned off1 = off0 + (unsigned)(NPB * KDIM * 4 / 2);    // second 8 KB half
    unsigned lds0 = lds_base + off0;
    unsigned lds1 = lds_base + off1;
    asm volatile("global_load_async_to_lds_b128 %0, %1, %2"
                 :: "v"(lds0), "v"(off0), "s"(gbase) : "memory");
    asm volatile("global_load_async_to_lds_b128 %0, %1, %2"
                 :: "v"(lds1), "v"(off1), "s"(gbase) : "memory");
    asm volatile("s_wait_asynccnt 0x0" ::: "memory");
  }
  __syncthreads();

  // Thread -> output mapping: k = t/16, d = (t%16)*4.
  // Per n-step the block covers R[b,n,0:32,0:64] = 8 KB, fully coalesced b128.
  const int k = t >> 4;
  const int d = (t & 15) << 2;

  const v4f* rp = (const v4f*)(R + ((((size_t)b * NDIM + n0) * KDIM + k) * DDIM + d));
  const int rstride = KDIM * DDIM / 4;  // 512 v4f per n-step

  v4f acc = (v4f)0.0f;
#pragma unroll 8
  for (int n = 0; n < NPB; ++n) {
    v4f r = __builtin_nontemporal_load(rp);   // global_load_b128 th:NT (1 GiB stream)
    float a = sA[n * KDIM + k];               // LDS broadcast (16 lanes same addr)
    acc += a * r;
    rp += rstride;
  }

  float* op = out + ((size_t)b * KDIM + k) * DDIM + d;
  __hip_atomic_fetch_add(op + 0, acc.x, __ATOMIC_RELAXED, __HIP_MEMORY_SCOPE_AGENT);
  __hip_atomic_fetch_add(op + 1, acc.y, __ATOMIC_RELAXED, __HIP_MEMORY_SCOPE_AGENT);
  __hip_atomic_fetch_add(op + 2, acc.z, __ATOMIC_RELAXED, __HIP_MEMORY_SCOPE_AGENT);
  __hip_atomic_fetch_add(op + 3, acc.w, __ATOMIC_RELAXED, __HIP_MEMORY_SCOPE_AGENT);
}

__global__ void zero_out_kernel(float* __restrict__ out, int n) {
  int i = blockIdx.x * blockDim.x + threadIdx.x;
  if (i < n) out[i] = 0.0f;
}

extern "C" void kernel_launch(void* const* d_in, const int* in_sizes, int n_in,
                              void* d_out, int out_size, void* d_ws, size_t ws_size,
                              hipStream_t stream) {
  const float* A = (const float*)d_in[0];  // [16, 8192, 32]
  const float* R = (const float*)d_in[1];  // [16, 8192, 32, 64]
  float* out = (float*)d_out;              // [16, 32, 64]

  zero_out_kernel<<<(out_size + 255) / 256, 256, 0, stream>>>(out, out_size);
  aggregate_kernel<<<BDIM * NSEG, THREADS, 0, stream>>>(A, R, out);
}